// MLPPredictor_18159121728021
// MI455X (gfx1250) — compile-verified
//
#include <hip/hip_runtime.h>

typedef float v2f __attribute__((ext_vector_type(2)));
typedef float v8f __attribute__((ext_vector_type(8)));

#define H1 128          // layer-1 output width
#define H2 64           // layer-2 output width
#define KIN 256         // layer-1 input width (concat src+dst)
#define LDS_STRIDE 132  // 128 + 4 pad floats -> conflict-free transpose staging

__global__ __launch_bounds__(128) void edge_mlp_kernel(
    const float* __restrict__ h,
    const int*   __restrict__ src_idx,
    const int*   __restrict__ dst_idx,
    const float* __restrict__ W1, const float* __restrict__ b1,
    const float* __restrict__ W2, const float* __restrict__ b2,
    const float* __restrict__ W3, const float* __restrict__ b3,
    float* __restrict__ out, int E)
{
    // 4 waves per block, each wave privately stages its 16x128 layer-1 tile.
    __shared__ float xstage[4 * 16 * LDS_STRIDE];

    const int lane = threadIdx.x & 31;
    const int wave = threadIdx.x >> 5;
    const int l15  = lane & 15;    // edge-in-tile (A rows / B cols)
    const int half = lane >> 4;    // K sub-offset selector for WMMA layout
    float* stage = xstage + wave * 16 * LDS_STRIDE;

    const int tileBase = (blockIdx.x * 4 + wave) * 16;

    // Clamp (no branch) so EXEC stays all-ones for WMMA on a ragged tail.
    int e = tileBase + l15;
    if (e > E - 1) e = E - 1;
    const int s = src_idx[e];
    const int d = dst_idx[e];
    const float* rowS = h + (long long)s * H1 + 2 * half;
    const float* rowD = h + (long long)d * H1 + 2 * half;

    // Per-lane bias / W3 preload (lane n handles output column n0 + l15).
    float bias1[8];
#pragma unroll
    for (int nt = 0; nt < 8; ++nt) bias1[nt] = b1[nt * 16 + l15];
    float bias2[4], w3v[4];
#pragma unroll
    for (int nt = 0; nt < 4; ++nt) {
        bias2[nt] = b2[nt * 16 + l15];
        w3v[nt]   = W3[nt * 16 + l15];
    }

    // ---------------- Layer 1: [16 x 256] @ W1^T -> [16 x 128] ----------------
    // Split the K=256 contraction into the src half (K 0..127) and dst half
    // (K 128..255) so each half has a single base pointer and the compiler can
    // fold the chunk offset into the load's immediate offset (no per-iteration
    // cndmask pointer selects).
    v8f acc[8] = {};
    const float* w1p = W1 + l15 * KIN + 2 * half;   // B operand: lane n -> W1 row n
    for (int c = 0; c < 32; ++c) {                  // src half: 32 K-chunks of 4
        v2f a = *(const v2f*)(rowS + 4 * c);        // A: K = 4c + 2*half + {0,1}
#pragma unroll
        for (int nt = 0; nt < 8; ++nt) {
            v2f bm = *(const v2f*)(w1p + nt * 16 * KIN + 4 * c);
            acc[nt] = __builtin_amdgcn_wmma_f32_16x16x4_f32(
                false, a, false, bm, (short)0, acc[nt], false, false);
        }
    }
    for (int c = 0; c < 32; ++c) {                  // dst half: 32 K-chunks of 4
        v2f a = *(const v2f*)(rowD + 4 * c);
#pragma unroll
        for (int nt = 0; nt < 8; ++nt) {
            v2f bm = *(const v2f*)(w1p + nt * 16 * KIN + 4 * (c + 32));
            acc[nt] = __builtin_amdgcn_wmma_f32_16x16x4_f32(
                false, a, false, bm, (short)0, acc[nt], false, false);
        }
    }

    // Bias + ReLU, stage to LDS transposed into WMMA A-layout for layer 2.
#pragma unroll
    for (int nt = 0; nt < 8; ++nt) {
#pragma unroll
        for (int r = 0; r < 8; ++r) {
            float x = fmaxf(acc[nt][r] + bias1[nt], 0.0f);
            // D layout: element (m = r + 8*half, n = nt*16 + l15)
            stage[(r + 8 * half) * LDS_STRIDE + nt * 16 + l15] = x;
        }
    }

    // ---------------- Layer 2: [16 x 128] @ W2^T -> [16 x 64] -----------------
    v8f acc2[4] = {};
    const float* w2p = W2 + l15 * H1 + 2 * half;
    const float* sp  = stage + l15 * LDS_STRIDE + 2 * half;  // A: row m = l15
    for (int c = 0; c < 32; ++c) {
        v2f a = *(const v2f*)(sp + 4 * c);
#pragma unroll
        for (int nt = 0; nt < 4; ++nt) {
            v2f bm = *(const v2f*)(w2p + nt * 16 * H1 + 4 * c);
            acc2[nt] = __builtin_amdgcn_wmma_f32_16x16x4_f32(
                false, a, false, bm, (short)0, acc2[nt], false, false);
        }
    }

    // ---------------- Layer 3: [16 x 64] @ W3^T + cross-lane reduce -----------
    float sred[8];
#pragma unroll
    for (int r = 0; r < 8; ++r) sred[r] = 0.0f;
#pragma unroll
    for (int nt = 0; nt < 4; ++nt) {
#pragma unroll
        for (int r = 0; r < 8; ++r) {
            float x = fmaxf(acc2[nt][r] + bias2[nt], 0.0f);
            sred[r] += x * w3v[nt];
        }
    }
    // Sum over the 16 N-lanes; masks 1..8 keep the two 16-lane halves separate
    // (half 0 holds edges m=0..7, half 1 holds m=8..15).
#pragma unroll
    for (int mask = 1; mask <= 8; mask <<= 1) {
#pragma unroll
        for (int r = 0; r < 8; ++r)
            sred[r] += __shfl_xor(sred[r], mask, 32);
    }

    if (l15 == 0) {
        const float bb3 = b3[0];
        const int m0 = tileBase + 8 * half;
#pragma unroll
        for (int r = 0; r < 8; ++r) {
            const int idx = m0 + r;
            if (idx < E) out[idx] = sred[r] + bb3;
        }
    }
}

extern "C" void kernel_launch(void* const* d_in, const int* in_sizes, int n_in,
                              void* d_out, int out_size, void* d_ws, size_t ws_size,
                              hipStream_t stream) {
    const float* h   = (const float*)d_in[0];
    const int*   src = (const int*)  d_in[1];
    const int*   dst = (const int*)  d_in[2];
    const float* W1  = (const float*)d_in[3];
    const float* b1  = (const float*)d_in[4];
    const float* W2  = (const float*)d_in[5];
    const float* b2  = (const float*)d_in[6];
    const float* W3  = (const float*)d_in[7];
    const float* b3  = (const float*)d_in[8];
    float* out = (float*)d_out;

    const int E = in_sizes[1];                 // number of edges
    const int blocks = (E + 63) / 64;          // 4 waves x 16 edges per block
    edge_mlp_kernel<<<blocks, 128, 0, stream>>>(h, src, dst, W1, b1, W2, b2,
                                                W3, b3, out, E);
}